// GAT_2740189135453
// MI455X (gfx1250) — compile-verified
//
#include <hip/hip_runtime.h>

typedef __attribute__((ext_vector_type(2))) float v2f;
typedef __attribute__((ext_vector_type(8))) float v8f;

static constexpr int NN      = 50000;
static constexpr int EE      = 800000;
static constexpr int IN_C    = 128;
static constexpr int HID     = 16;
static constexpr int OUT_C   = 64;
static constexpr int HEADS   = 4;
static constexpr int HF      = 64;      // heads*F == out width for every layer
static constexpr float NEG_SLOPE = 0.2f;

// ---- monotone f32 <-> u32 ordered key (for atomic max incl. negatives) ----
__device__ __forceinline__ unsigned f32_key(float f) {
  unsigned u = __float_as_uint(f);
  return (u & 0x80000000u) ? ~u : (u | 0x80000000u);
}
__device__ __forceinline__ float key_f32(unsigned k) {
  unsigned u = (k & 0x80000000u) ? (k ^ 0x80000000u) : ~k;
  return __uint_as_float(u);
}
// f32_key(-inf) = ~0xFF800000 = 0x007FFFFF
static constexpr unsigned NEG_INF_KEY = 0x007FFFFFu;

__device__ __forceinline__ float leaky(float x) {
  return x > 0.f ? x : NEG_SLOPE * x;
}

// ------------------------------------------------------------------
// f32 GEMM: C[M,NCOL] = A[M,K] @ B[K,NCOL], one wave per 16x16 tile,
// using V_WMMA_F32_16X16X4_F32 (full fp32 precision, K-step 4).
// K, NCOL are compile-time so the K-loop unrolls evenly (no scalar
// remainder path) into pipelined load + wmma groups.
// A 16x4 layout: lane l holds row m=l%16, K pair {0,1} (lanes 0-15)
// or {2,3} (lanes 16-31) in 2 VGPRs. B mirrored (col n=l%16).
// C/D: VGPR r -> row r (lanes 0-15) / r+8 (lanes 16-31), col = l%16.
// ------------------------------------------------------------------
template<int K, int NCOL>
__global__ void gat_gemm_wmma(const float* __restrict__ A,
                              const float* __restrict__ B,
                              float* __restrict__ C,
                              int M) {
  const int lane  = threadIdx.x & 31;
  const int wave  = (blockIdx.x * blockDim.x + threadIdx.x) >> 5;
  constexpr int ntn = NCOL >> 4;
  const int tm    = wave / ntn;
  const int tn    = wave - tm * ntn;
  if (tm * 16 >= M) return;

  const int lm    = lane & 15;
  const int khalf = (lane >> 4) << 1;            // 0 or 2
  const float* ap = A + (size_t)(tm * 16 + lm) * K + khalf;
  const float* bp = B + (size_t)khalf * NCOL + tn * 16 + lm;

  v8f c = {};
#pragma unroll 8
  for (int k = 0; k < K; k += 4) {
    v2f a, b;
    a.x = ap[0];      a.y = ap[1];
    b.x = bp[0];      b.y = bp[NCOL];
    ap += 4;
    bp += 4 * NCOL;
    c = __builtin_amdgcn_wmma_f32_16x16x4_f32(false, a, false, b,
                                              (short)0, c, false, false);
  }

  const int mbase = tm * 16 + ((lane >> 4) << 3);  // +8 for upper lane half
  float* cp = C + (size_t)mbase * NCOL + tn * 16 + lm;
#pragma unroll
  for (int r = 0; r < 8; ++r) cp[(size_t)r * NCOL] = c[r];
}

// ---- alpha_s / alpha_d: per (node, head) dot of F features ----
template<int H, int F>
__global__ void gat_alpha(const float* __restrict__ h,
                          const float* __restrict__ a_src,
                          const float* __restrict__ a_dst,
                          float* __restrict__ as_, float* __restrict__ ad_) {
  int i = blockIdx.x * blockDim.x + threadIdx.x;
  if (i >= NN * H) return;
  int n = i / H, hh = i - n * H;
  const float* hp  = h + (size_t)n * (H * F) + hh * F;
  const float* sp  = a_src + hh * F;
  const float* dp  = a_dst + hh * F;
  float s = 0.f, d = 0.f;
#pragma unroll
  for (int f = 0; f < F; ++f) { s += hp[f] * sp[f]; d += hp[f] * dp[f]; }
  as_[i] = s; ad_[i] = d;
}

// ---- init accumulators for a layer ----
template<int H>
__global__ void gat_init(float* __restrict__ acc, unsigned* __restrict__ mkey,
                         float* __restrict__ denom) {
  int i = blockIdx.x * blockDim.x + threadIdx.x;
  if (i < NN * H) { mkey[i] = NEG_INF_KEY; denom[i] = 0.f; }
  if (i < NN * HF) acc[i] = 0.f;
}

// ---- edge pass 1: segment max of leaky(as[src]+ad[dst]) per (dst, head) ----
template<int H>
__global__ void gat_edge_max(const int* __restrict__ esrc, const int* __restrict__ edst,
                             const float* __restrict__ as_, const float* __restrict__ ad_,
                             unsigned* __restrict__ mkey) {
  int i = blockIdx.x * blockDim.x + threadIdx.x;
  if (i >= (EE + NN) * H) return;
  int e = i / H, hh = i - e * H;
  int s, d;
  if (e < EE) { s = esrc[e]; d = edst[e]; } else { s = d = e - EE; }
  float ev = leaky(as_[s * H + hh] + ad_[d * H + hh]);
  atomicMax(&mkey[d * H + hh], f32_key(ev));
}

// ---- edge pass 2: w = exp(e - m[dst]); denom[dst] += w; acc[dst] += w*h[src]
// 64 threads per edge (one per output feature) => 256B coalesced gather of
// h[src] and contiguous atomic range on acc[dst]; lane f%F==0 adds denom. ----
template<int H, int F>
__global__ void gat_edge_aggr(const int* __restrict__ esrc, const int* __restrict__ edst,
                              const float* __restrict__ as_, const float* __restrict__ ad_,
                              const unsigned* __restrict__ mkey,
                              float* __restrict__ denom,
                              const float* __restrict__ hfeat,
                              float* __restrict__ acc) {
  int i = blockIdx.x * blockDim.x + threadIdx.x;
  if (i >= (EE + NN) * HF) return;
  int e = i / HF, f = i - e * HF, hh = f / F;
  int s, d;
  if (e < EE) { s = esrc[e]; d = edst[e]; } else { s = d = e - EE; }
  float ev = leaky(as_[s * H + hh] + ad_[d * H + hh]);
  float w  = __expf(ev - key_f32(mkey[d * H + hh]));
  if ((f & (F - 1)) == 0) atomicAdd(&denom[d * H + hh], w);
  atomicAdd(&acc[(size_t)d * HF + f], hfeat[(size_t)s * HF + f] * w);
}

// ---- finalize: divide by denom, add bias, optional ELU ----
template<int H, int F, bool ELU>
__global__ void gat_finalize(const float* __restrict__ acc,
                             const float* __restrict__ denom,
                             const float* __restrict__ bias,
                             float* __restrict__ out) {
  int i = blockIdx.x * blockDim.x + threadIdx.x;
  if (i >= NN * HF) return;
  int n = i / HF, f = i - n * HF, hh = f / F;
  float v = acc[i] / (denom[n * H + hh] + 1e-16f) + bias[f];
  if (ELU) v = v > 0.f ? v : (__expf(v) - 1.f);
  out[i] = v;
}

static inline int cdiv(long long a, int b) { return (int)((a + b - 1) / b); }

extern "C" void kernel_launch(void* const* d_in, const int* in_sizes, int n_in,
                              void* d_out, int out_size, void* d_ws, size_t ws_size,
                              hipStream_t stream) {
  const float* x   = (const float*)d_in[0];
  const int* esrc  = (const int*)d_in[1];
  const int* edst  = esrc + EE;
  const float* W0  = (const float*)d_in[2];
  const float* as0 = (const float*)d_in[3];
  const float* ad0 = (const float*)d_in[4];
  const float* b0  = (const float*)d_in[5];
  const float* W1  = (const float*)d_in[6];
  const float* as1 = (const float*)d_in[7];
  const float* ad1 = (const float*)d_in[8];
  const float* b1  = (const float*)d_in[9];
  const float* W2  = (const float*)d_in[10];
  const float* as2 = (const float*)d_in[11];
  const float* ad2 = (const float*)d_in[12];
  const float* b2  = (const float*)d_in[13];
  float* out = (float*)d_out;

  // workspace carve (all f32): hfeat | acc | xbuf | as | ad | mkey | denom
  float* ws      = (float*)d_ws;
  float* hfeat   = ws;                     // NN*64
  float* acc     = hfeat + (size_t)NN * HF;
  float* xbuf    = acc   + (size_t)NN * HF;
  float* as_     = xbuf  + (size_t)NN * HF;
  float* ad_     = as_   + (size_t)NN * HEADS;
  unsigned* mkey = (unsigned*)(ad_ + (size_t)NN * HEADS);
  float* denom   = (float*)(mkey + (size_t)NN * HEADS);

  const int TB = 256;
  const long long nEdgeH4 = (long long)(EE + NN) * HEADS;
  const long long nEdgeH1 = (long long)(EE + NN) * 1;
  const long long nEdgeHF = (long long)(EE + NN) * HF;
  const long long nNodeHF = (long long)NN * HF;
  const int gemmThreads = (NN / 16) * (HF / 16) * 32;   // one wave per tile

  // ---------------- Layer 0: [N,128] -> heads=4, F=16, concat, ELU --------
  {
    gat_gemm_wmma<IN_C, HF><<<cdiv(gemmThreads, TB), TB, 0, stream>>>(
        x, W0, hfeat, NN);
    gat_alpha<HEADS, HID><<<cdiv((long long)NN * HEADS, TB), TB, 0, stream>>>(
        hfeat, as0, ad0, as_, ad_);
    gat_init<HEADS><<<cdiv(nNodeHF, TB), TB, 0, stream>>>(acc, mkey, denom);
    gat_edge_max<HEADS><<<cdiv(nEdgeH4, TB), TB, 0, stream>>>(
        esrc, edst, as_, ad_, mkey);
    gat_edge_aggr<HEADS, HID><<<cdiv(nEdgeHF, TB), TB, 0, stream>>>(
        esrc, edst, as_, ad_, mkey, denom, hfeat, acc);
    gat_finalize<HEADS, HID, true><<<cdiv(nNodeHF, TB), TB, 0, stream>>>(
        acc, denom, b0, xbuf);
  }

  // ---------------- Layer 1: [N,64] -> heads=4, F=16, concat, ELU ---------
  {
    gat_gemm_wmma<HF, HF><<<cdiv(gemmThreads, TB), TB, 0, stream>>>(
        xbuf, W1, hfeat, NN);
    gat_alpha<HEADS, HID><<<cdiv((long long)NN * HEADS, TB), TB, 0, stream>>>(
        hfeat, as1, ad1, as_, ad_);
    gat_init<HEADS><<<cdiv(nNodeHF, TB), TB, 0, stream>>>(acc, mkey, denom);
    gat_edge_max<HEADS><<<cdiv(nEdgeH4, TB), TB, 0, stream>>>(
        esrc, edst, as_, ad_, mkey);
    gat_edge_aggr<HEADS, HID><<<cdiv(nEdgeHF, TB), TB, 0, stream>>>(
        esrc, edst, as_, ad_, mkey, denom, hfeat, acc);
    gat_finalize<HEADS, HID, true><<<cdiv(nNodeHF, TB), TB, 0, stream>>>(
        acc, denom, b1, xbuf);
  }

  // ---------------- Layer 2: [N,64] -> heads=1, F=64, mean (=identity) ----
  {
    gat_gemm_wmma<HF, OUT_C><<<cdiv(gemmThreads, TB), TB, 0, stream>>>(
        xbuf, W2, hfeat, NN);
    gat_alpha<1, OUT_C><<<cdiv((long long)NN, TB), TB, 0, stream>>>(
        hfeat, as2, ad2, as_, ad_);
    gat_init<1><<<cdiv(nNodeHF, TB), TB, 0, stream>>>(acc, mkey, denom);
    gat_edge_max<1><<<cdiv(nEdgeH1, TB), TB, 0, stream>>>(
        esrc, edst, as_, ad_, mkey);
    gat_edge_aggr<1, OUT_C><<<cdiv(nEdgeHF, TB), TB, 0, stream>>>(
        esrc, edst, as_, ad_, mkey, denom, hfeat, acc);
    gat_finalize<1, OUT_C, false><<<cdiv(nNodeHF, TB), TB, 0, stream>>>(
        acc, denom, b2, out);
  }
}